// FlexTree_26379689132517
// MI455X (gfx1250) — compile-verified
//
#include <hip/hip_runtime.h>
#include <hip/hip_bf16.h>

typedef __attribute__((ext_vector_type(2))) float v2f;
typedef __attribute__((ext_vector_type(8))) float v8f;

#define N_ROWS 1048576
#define F_DIM  64
#define C_DIM  8
#define D_DIM  4
#define S_COLS 16   // pad 8 real children columns to 16 for the WMMA N dim

// One block = 256 threads = 8 wave32s; each wave computes a 16x16 z-tile via
// 16x V_WMMA_F32_16X16X4_F32 over K=64, then does the elementwise gaussian +
// 8-wide cross-lane reduction + outer gaussian.
__global__ __launch_bounds__(256) void FlexTree_wmma_kernel(
    const float* __restrict__ X,        // (N, 64)
    const int*   __restrict__ leaf_idx, // (8, 4)
    const float* __restrict__ W_child,  // (8, 4)
    const float* __restrict__ w_root,   // (8,)
    const float* __restrict__ a_ptr,    // scalar
    const float* __restrict__ b_ptr,    // scalar
    float*       __restrict__ y)        // (N, 1)
{
    // Scatter matrix S[f][c] = sum_d W_child[c,d] * (leaf_idx[c,d]==f), padded to 16 cols.
    __shared__ float S[F_DIM * S_COLS];   // 4 KB

    const int tid = threadIdx.x;

    // Zero S (1024 floats / 256 threads).
    #pragma unroll
    for (int i = 0; i < 4; ++i)
        S[tid + i * 256] = 0.0f;
    __syncthreads();

    // Scatter-accumulate the 32 (c,d) weights. atomicAdd handles duplicate
    // leaf indices within a child exactly like the reference's einsum.
    if (tid < C_DIM * D_DIM) {
        const int c = tid >> 2;
        const int d = tid & 3;
        const int f = leaf_idx[c * D_DIM + d];
        atomicAdd(&S[f * S_COLS + c], W_child[c * D_DIM + d]);
    }
    __syncthreads();

    const float a     = a_ptr[0];
    const float inv_b = 1.0f / b_ptr[0];

    const int wave  = tid >> 5;        // 0..7
    const int lane  = tid & 31;        // wave32 lane
    const int laneN = lane & 15;       // N column (C/D) or M row (A)
    const int hi    = lane >> 4;       // lane half selects K {0,1} vs {2,3}

    const int rowBase = (blockIdx.x * 8 + wave) * 16;

    // A-fragment source: lane holds X[rowBase+laneN][4*kb + 2*hi + {0,1}]
    const float* Aptr = X + (size_t)(rowBase + laneN) * F_DIM + 2 * hi;

    v8f acc = {};  // z tile accumulator (16x16 f32, 8 VGPRs)

    #pragma unroll
    for (int kb = 0; kb < 16; ++kb) {
        // 64-bit coalesced-per-pair global load of the A fragment.
        v2f afrag = *(const v2f*)(Aptr + kb * 4);

        // B fragment from LDS: VGPR v holds K = 4*kb + v + 2*hi, N = laneN.
        const int kRow = kb * 4 + 2 * hi;
        v2f bfrag;
        bfrag.x = S[(kRow + 0) * S_COLS + laneN];
        bfrag.y = S[(kRow + 1) * S_COLS + laneN];

        // D = A x B + C, full fp32.  (neg_a, A, neg_b, B, c_mod, C, reuse_a, reuse_b)
        acc = __builtin_amdgcn_wmma_f32_16x16x4_f32(
            false, afrag, false, bfrag, (short)0, acc, false, false);
    }

    // Child weight for this lane's column; padded columns (N>=8) weigh 0,
    // so their finite g(0) contributes nothing (no NaN/Inf possible).
    const float wr = (laneN < C_DIM) ? w_root[laneN] : 0.0f;

    #pragma unroll
    for (int v = 0; v < 8; ++v) {
        // acc[v]: lanes 0-15 -> row rowBase+v, lanes 16-31 -> row rowBase+v+8.
        const float z  = acc[v];
        const float t  = (z - a) * inv_b;
        float       hw = __expf(-t * t) * wr;

        // Sum over the 8 child columns: xor-reduce within groups of 8 lanes.
        hw += __shfl_xor(hw, 1, 32);
        hw += __shfl_xor(hw, 2, 32);
        hw += __shfl_xor(hw, 4, 32);

        const float u  = (hw - a) * inv_b;
        const float yv = __expf(-u * u);

        // Lane 0 holds row rowBase+v, lane 16 holds row rowBase+v+8.
        if (laneN == 0)
            y[rowBase + v + 8 * hi] = yv;
    }
}

extern "C" void kernel_launch(void* const* d_in, const int* in_sizes, int n_in,
                              void* d_out, int out_size, void* d_ws, size_t ws_size,
                              hipStream_t stream) {
    const float* X        = (const float*)d_in[0];
    const int*   leaf_idx = (const int*)d_in[1];
    const float* W_child  = (const float*)d_in[2];
    const float* w_root   = (const float*)d_in[3];
    const float* a_ptr    = (const float*)d_in[4];
    const float* b_ptr    = (const float*)d_in[5];
    float*       y        = (float*)d_out;

    // 1048576 rows / (8 waves * 16 rows) = 8192 blocks of 256 threads.
    dim3 grid(N_ROWS / 128);
    dim3 block(256);
    FlexTree_wmma_kernel<<<grid, block, 0, stream>>>(
        X, leaf_idx, W_child, w_root, a_ptr, b_ptr, y);
}